// Model_50989851738461
// MI455X (gfx1250) — compile-verified
//
#include <hip/hip_runtime.h>
#include <stdint.h>

// ---------------------------------------------------------------------------
// SMPL-X kinematic-chain global rotations, MI455X (gfx1250).
// Bandwidth-bound (415 MB moved, ~0.3 GFLOP): stream tiles with the CDNA5
// Tensor Data Mover (TENSOR_LOAD_TO_LDS / TENSOR_STORE_FROM_LDS), compute the
// dependent 3x3 chain per batch element out of LDS (bank-conflict-free).
// ---------------------------------------------------------------------------

typedef __attribute__((ext_vector_type(4))) unsigned int v4u;
typedef __attribute__((ext_vector_type(8))) int          v8i;
typedef __attribute__((ext_vector_type(4))) int          v4i;

#define NJ      22
#define ELEM_F  (NJ * 9)          // 198 floats per batch element
#define TPB     64                // threads per block == batch elems per tile
#define TILE_F  (TPB * ELEM_F)    // 12672 floats = 49.5 KB LDS per block

// Joint processing order grouped into parent->child runs so the parent's
// global rotation stays in registers within a run. {joint, parent}.
constexpr int ORDER[21][2] = {
  {1,0},{4,1},{7,4},{10,7},
  {2,0},{5,2},{8,5},{11,8},
  {3,0},{6,3},{9,6},
  {12,9},{15,12},
  {13,9},{16,13},{18,16},{20,18},
  {14,9},{17,14},{19,17},{21,19}
};

// --- Tensor DMA descriptor (D#) builders ------------------------------------
// Group 0 (128b): [1:0]=count=1, [63:32]=lds_addr, [120:64]=global_addr,
//                 [127:126]=type=2 ("image").
__device__ inline v4u tdm_group0(uint32_t lds_addr, uint64_t gaddr) {
  v4u g;
  g[0] = 1u;                       // count=1, user descriptor, gather off
  g[1] = lds_addr;                 // LDS byte offset (wave-relative)
  g[2] = (uint32_t)gaddr;          // global_addr[31:0]
  g[3] = ((uint32_t)(gaddr >> 32) & 0x01FFFFFFu) | (2u << 30); // [56:32] | type=2
  return g;
}

// Group 1 (256b): 1-D copy of TILE_F dwords. data_size code 2 = 4 bytes.
// tensor_dim0 set huge so no OOB clipping (grid divides evenly, all tiles full).
__device__ inline v8i tdm_group1() {
  const uint32_t tdim0 = 0x7FFFFFFFu;
  v8i g;
  g[0] = (int)(2u << 16);                      // wg_mask=0 | data_size=4B
  g[1] = (int)((tdim0 & 0xFFFFu) << 16);       // barrier_addr=0 | tensor_dim0[15:0]
  g[2] = (int)((tdim0 >> 16) | (1u << 16));    // tensor_dim0[31:16] | tensor_dim1.lo=1
  g[3] = (int)(((uint32_t)TILE_F) << 16);      // tensor_dim1.hi=0 | tile_dim0=TILE_F
  g[4] = 0;                                    // tile_dim1=0, tile_dim2=0 (1-D tile)
  g[5] = (int)TILE_F;                          // tensor_dim0_stride[31:0]
  g[6] = 0;                                    // stride0.hi | stride1.lo
  g[7] = 0;                                    // stride1.hi
  return g;
}

__global__ __launch_bounds__(TPB) void
chain_rot_kernel(const float* __restrict__ in, float* __restrict__ out) {
  __shared__ float tile[TILE_F];

  const uint32_t lds_base = (uint32_t)(uintptr_t)&tile[0]; // low 32b = LDS offset
  const size_t   tile_off = (size_t)blockIdx.x * (size_t)TILE_F;

  // --- DMA the whole tile Global -> LDS (one TDM op, issued by wave 0) ---
  if (threadIdx.x < 32) {
    v4u g0 = tdm_group0(lds_base, (uint64_t)(uintptr_t)(in + tile_off));
    v8i g1 = tdm_group1();
    v4i z4 = {0, 0, 0, 0};
    v8i z8 = {0, 0, 0, 0, 0, 0, 0, 0};
    __builtin_amdgcn_tensor_load_to_lds(g0, g1, z4, z4, z8, 0);
    __builtin_amdgcn_s_wait_tensorcnt(0);
  }
  __syncthreads();

  // --- Per-thread dependent 3x3 chain, in place in LDS ---
  // Lane stride 198 dwords ≡ 6 (mod 64 banks): 32 lanes -> 32 distinct banks.
  const int base = (int)threadIdx.x * ELEM_F;
  float g[9] = {0}, r[9], m[9];
#pragma unroll
  for (int k = 0; k < 21; ++k) {
    const int  j      = ORDER[k][0];
    const int  p      = ORDER[k][1];
    const bool reload = (k == 0) || (ORDER[k - 1][0] != p);
    const int  pb = base + p * 9;
    const int  jb = base + j * 9;
    if (reload) {
#pragma unroll
      for (int e = 0; e < 9; ++e) g[e] = tile[pb + e];
    }
#pragma unroll
    for (int e = 0; e < 9; ++e) r[e] = tile[jb + e];
#pragma unroll
    for (int rr = 0; rr < 3; ++rr) {
#pragma unroll
      for (int cc = 0; cc < 3; ++cc) {
        m[rr * 3 + cc] = fmaf(g[rr * 3 + 0], r[0 + cc],
                         fmaf(g[rr * 3 + 1], r[3 + cc],
                              g[rr * 3 + 2] * r[6 + cc]));
      }
    }
#pragma unroll
    for (int e = 0; e < 9; ++e) { tile[jb + e] = m[e]; g[e] = m[e]; }
  }
  __syncthreads();

  // --- DMA the tile LDS -> Global output (one TDM op, wave 0) ---
  if (threadIdx.x < 32) {
    v4u g0 = tdm_group0(lds_base, (uint64_t)(uintptr_t)(out + tile_off));
    v8i g1 = tdm_group1();
    v4i z4 = {0, 0, 0, 0};
    v8i z8 = {0, 0, 0, 0, 0, 0, 0, 0};
    __builtin_amdgcn_tensor_store_from_lds(g0, g1, z4, z4, z8, 0);
    __builtin_amdgcn_s_wait_tensorcnt(0);
  }
}

extern "C" void kernel_launch(void* const* d_in, const int* in_sizes, int n_in,
                              void* d_out, int out_size, void* d_ws, size_t ws_size,
                              hipStream_t stream) {
  const float* in  = (const float*)d_in[0];
  float*       out = (float*)d_out;
  const int nbatch  = in_sizes[0] / ELEM_F;   // 262144
  const int nblocks = nbatch / TPB;           // 4096 full tiles
  chain_rot_kernel<<<nblocks, TPB, 0, stream>>>(in, out);
}